// Loss_21835613733077
// MI455X (gfx1250) — compile-verified
//
#include <hip/hip_runtime.h>
#include <cstdint>
#include <cstddef>

#define TPB   256
#define CDIM  10
#define TSAMP 25

// ---- CDNA5 async global->LDS path (guarded; falls back cleanly) ----------
#if defined(__gfx1250__) && __has_builtin(__builtin_amdgcn_global_load_async_to_lds_b64)
#define HAVE_ASYNC 1
#else
#define HAVE_ASYNC 0
#endif

#if HAVE_ASYNC
#define AS_GLOBAL __attribute__((address_space(1)))
#define AS_LOCAL  __attribute__((address_space(3)))
typedef int v2i __attribute__((ext_vector_type(2)));
#if __has_builtin(__builtin_amdgcn_s_wait_asynccnt)
#define WAIT_ASYNC(n) __builtin_amdgcn_s_wait_asynccnt(n)
#else
#define WAIT_ASYNC(n) asm volatile("s_wait_asynccnt %0" ::"i"(n) : "memory")
#endif
// 64-bit async copy: global (addrspace 1) -> LDS (addrspace 3)
#define ASYNC_COPY_B64(gp, lp)                                         \
    __builtin_amdgcn_global_load_async_to_lds_b64(                     \
        (AS_GLOBAL v2i*)(gp), (AS_LOCAL v2i*)(lp), 0, 0)
#endif

__global__ __launch_bounds__(TPB) void mc_nll_partial(
    const float* __restrict__ mu,
    const float* __restrict__ ls,
    const int*   __restrict__ y,
    const float* __restrict__ eps,
    float* __restrict__ partial,
    int B)
{
    const int tid = threadIdx.x;
    const int b   = blockIdx.x * TPB + tid;
    const bool active = (b < B);

#if HAVE_ASYNC
    // Double-buffered LDS tile: row stride 10 dwords -> 32 lanes hit 32
    // distinct banks (gcd(10,64)=2, period 32): conflict-free ds_load_b64.
    __shared__ float tile[2][TPB * CDIM];

    // Issue the t=0 copies first so they are in flight while we set up
    // the per-thread constants below (5 x B64 = 40 bytes per lane).
    if (active) {
        const float* g = eps + (size_t)b * CDIM;
        float*       l = &tile[0][tid * CDIM];
#pragma unroll
        for (int j = 0; j < 5; ++j)
            ASYNC_COPY_B64(g + 2 * j, l + 2 * j);
    }
#endif

    // Per-thread constants: mu row and sigma row (sigma = exp(0.5*log_sigma2))
    float muv[CDIM], sg[CDIM];
    int yb = 0;
    if (active) {
        const float* mrow = mu + (size_t)b * CDIM;
        const float* srow = ls + (size_t)b * CDIM;
#pragma unroll
        for (int c = 0; c < CDIM; ++c) {
            muv[c] = mrow[c];
            sg[c]  = __expf(0.5f * srow[c]);
        }
        yb = y[b];
    }

    // Online logsumexp over the T samples of a_t = logit_y - lse_c(logits)
    float m = -INFINITY, s = 0.0f;

#if HAVE_ASYNC
    for (int t = 0; t < TSAMP; ++t) {
        // 1) Wait for batch t (the only outstanding batch) to land in LDS.
        WAIT_ASYNC(0);

        // 2) Drain this thread's row into registers (5 x ds_load_b64).
        float ev[CDIM];
        if (active) {
            const float2* e2 = (const float2*)&tile[t & 1][tid * CDIM];
#pragma unroll
            for (int j = 0; j < 5; ++j) {
                float2 p = e2[j];
                ev[2 * j] = p.x; ev[2 * j + 1] = p.y;
            }
        }

        // 3) Issue batch t+1 into the other buffer; it overlaps step 4.
        if (t + 1 < TSAMP && active) {
            const float* g = eps + ((size_t)(t + 1) * B + b) * CDIM;
            float*       l = &tile[(t + 1) & 1][tid * CDIM];
#pragma unroll
            for (int j = 0; j < 5; ++j)
                ASYNC_COPY_B64(g + 2 * j, l + 2 * j);
        }

        // 4) Compute a_t and fold into the running logsumexp.
        if (active) {
            float li[CDIM];
            float rm = -INFINITY, ly = 0.0f;
#pragma unroll
            for (int c = 0; c < CDIM; ++c) {
                float v = fmaf(sg[c], ev[c], muv[c]);
                li[c] = v;
                rm = fmaxf(rm, v);
                ly = (c == yb) ? v : ly;
            }
            float se = 0.0f;
#pragma unroll
            for (int c = 0; c < CDIM; ++c) se += __expf(li[c] - rm);
            float a = ly - (rm + __logf(se));
            if (a > m) { s = s * __expf(m - a) + 1.0f; m = a; }
            else       { s += __expf(a - m); }
        }
    }
#else
    // Fallback: direct float2 streaming loads (rows are 8B-aligned: b*40).
    for (int t = 0; t < TSAMP; ++t) {
        if (active) {
            const float2* g2 = (const float2*)(eps + ((size_t)t * B + b) * CDIM);
            float ev[CDIM];
#pragma unroll
            for (int j = 0; j < 5; ++j) {
                float2 p = g2[j];
                ev[2 * j] = p.x; ev[2 * j + 1] = p.y;
            }
            float li[CDIM];
            float rm = -INFINITY, ly = 0.0f;
#pragma unroll
            for (int c = 0; c < CDIM; ++c) {
                float v = fmaf(sg[c], ev[c], muv[c]);
                li[c] = v;
                rm = fmaxf(rm, v);
                ly = (c == yb) ? v : ly;
            }
            float se = 0.0f;
#pragma unroll
            for (int c = 0; c < CDIM; ++c) se += __expf(li[c] - rm);
            float a = ly - (rm + __logf(se));
            if (a > m) { s = s * __expf(m - a) + 1.0f; m = a; }
            else       { s += __expf(a - m); }
        }
    }
#endif

    float picked = active ? (m + __logf(s) - __logf((float)TSAMP)) : 0.0f;

    // Deterministic in-block tree reduction of picked.
    __shared__ float red[TPB];
    red[tid] = picked;
    __syncthreads();
#pragma unroll
    for (int off = TPB / 2; off > 0; off >>= 1) {
        if (tid < off) red[tid] += red[tid + off];
        __syncthreads();
    }
    if (tid == 0) partial[blockIdx.x] = red[0];
}

__global__ __launch_bounds__(TPB) void mc_nll_finish(
    const float* __restrict__ partial, int nPartial,
    float* __restrict__ out, float invB)
{
    const int tid = threadIdx.x;
    float s = 0.0f;
    for (int i = tid; i < nPartial; i += TPB) s += partial[i];  // fixed order
    __shared__ float red[TPB];
    red[tid] = s;
    __syncthreads();
#pragma unroll
    for (int off = TPB / 2; off > 0; off >>= 1) {
        if (tid < off) red[tid] += red[tid + off];
        __syncthreads();
    }
    if (tid == 0) *out = -red[0] * invB;   // -mean(picked)
}

extern "C" void kernel_launch(void* const* d_in, const int* in_sizes, int n_in,
                              void* d_out, int out_size, void* d_ws, size_t ws_size,
                              hipStream_t stream) {
    const float* mu  = (const float*)d_in[0];   // [B, C] f32
    const float* ls  = (const float*)d_in[1];   // [B, C] f32
    const int*   y   = (const int*)d_in[2];     // [B] int
    const float* eps = (const float*)d_in[3];   // [T, B, C] f32

    const int B = in_sizes[0] / CDIM;
    const int nBlocks = (B + TPB - 1) / TPB;    // 2048 for B = 524288

    float* partial = (float*)d_ws;              // nBlocks floats of scratch

    mc_nll_partial<<<nBlocks, TPB, 0, stream>>>(mu, ls, y, eps, partial, B);
    mc_nll_finish<<<1, TPB, 0, stream>>>(partial, nBlocks, (float*)d_out,
                                         1.0f / (float)B);
}